// TIP_47751446397028
// MI455X (gfx1250) — compile-verified
//
#include <hip/hip_runtime.h>

#define N_PROT   20000
#define N_DRUG   4000
#define IN_DRUG  4000
#define IN_PROT  64
#define N_ET     128
#define N_BASE   32
#define N_EMBED  48
#define PD_DIM   16
#define H1       32
#define H2       16
#define RGCN_IN  64   // N_EMBED + PD_DIM
#define E_PP     800000
#define E_DP     400000
#define E_DD     1000000

typedef float v2f __attribute__((ext_vector_type(2)));
typedef float v8f __attribute__((ext_vector_type(8)));

// ---------------------------------------------------------------------------
// WMMA f32 GEMM (optionally batched over blockIdx.z):
//   C[z] [M x N] = A [M x K] * B[z] [K x N]      (row-major)
// grid = (M/16, N/16, NB), block = 32 (one full wave -> EXEC all ones).
// A frag (16x4 f32): lanes 0-15 hold K={k,k+1}, lanes 16-31 hold K={k+2,k+3},
// row = lane%16 (loaded as one float2). B frag symmetric with col = lane%16.
// C/D: VGPR r holds row r + 8*(lane/16), col = lane%16.
// ---------------------------------------------------------------------------
__global__ void gemm_wmma(const float* __restrict__ A, const float* __restrict__ B,
                          float* __restrict__ C, int K, int N,
                          long bStride, long cStride)
{
  const float* __restrict__ Bz = B + blockIdx.z * bStride;
  float* __restrict__       Cz = C + blockIdx.z * cStride;
  const int lane = threadIdx.x;
  const int half = lane >> 4;
  const int lp   = lane & 15;
  const int row  = blockIdx.x * 16 + lp;
  const int col  = blockIdx.y * 16 + lp;
  v8f c = {0.f, 0.f, 0.f, 0.f, 0.f, 0.f, 0.f, 0.f};
  for (int k = 0; k < K; k += 4) {
    const int kb = k + 2 * half;
    v2f a = *(const v2f*)(A + row * K + kb);   // 8B-aligned: K, kb even
    v2f b;
    b.x = Bz[kb * N + col];
    b.y = Bz[(kb + 1) * N + col];
    c = __builtin_amdgcn_wmma_f32_16x16x4_f32(false, a, false, b,
                                              (short)0, c, false, false);
  }
  const int m0 = blockIdx.x * 16 + 8 * half;
#pragma unroll
  for (int r = 0; r < 8; ++r)
    Cz[(m0 + r) * N + col] = c[r];
}

// ---------------------------------------------------------------------------
// Elementwise / scatter helpers
// ---------------------------------------------------------------------------
__global__ void k_fill(float* __restrict__ p, float v, int n) {
  int i = blockIdx.x * blockDim.x + threadIdx.x;
  if (i < n) p[i] = v;
}

__global__ void k_rsqrt(const float* __restrict__ a, float* __restrict__ b, int n) {
  int i = blockIdx.x * blockDim.x + threadIdx.x;
  if (i < n) b[i] = rsqrtf(a[i]);
}

// cnt[dst[i]-base] += 1
__global__ void k_count(const int* __restrict__ dst, float* __restrict__ cnt,
                        int n, int base) {
  int i = blockIdx.x * blockDim.x + threadIdx.x;
  if (i < n) atomicAdd(&cnt[dst[i] - base], 1.0f);
}

// agg = H * dis^2 + bias   (C = 1<<shift channels)
__global__ void k_gcn_init(const float* __restrict__ H, const float* __restrict__ dis,
                           const float* __restrict__ bias, float* __restrict__ agg,
                           int total, int shift) {
  int i = blockIdx.x * blockDim.x + threadIdx.x;
  if (i >= total) return;
  int node = i >> shift;
  int c    = i & ((1 << shift) - 1);
  float d  = dis[node];
  agg[i]   = H[i] * d * d + bias[c];
}

// agg[dst] += H[src] * dis[src]*dis[dst]   (thread = edge*channel)
__global__ void k_gcn_scatter(const float* __restrict__ H, const float* __restrict__ dis,
                              const int* __restrict__ src, const int* __restrict__ dst,
                              float* __restrict__ agg, int E, int shift) {
  int i = blockIdx.x * blockDim.x + threadIdx.x;
  if (i >= (E << shift)) return;
  int e = i >> shift;
  int c = i & ((1 << shift) - 1);
  int s = src[e], d = dst[e];
  atomicAdd(&agg[(d << shift) + c], H[(s << shift) + c] * dis[s] * dis[d]);
}

__global__ void k_relu(float* __restrict__ p, int n) {
  int i = blockIdx.x * blockDim.x + threadIdx.x;
  if (i < n) p[i] = fmaxf(p[i], 0.f);
}

// hierarchy: aggD[(dst-N_PROT)*16+c] += xp2[src*16+c]
__global__ void k_dp_scatter(const float* __restrict__ xp2, const int* __restrict__ src,
                             const int* __restrict__ dst, float* __restrict__ aggD) {
  int i = blockIdx.x * blockDim.x + threadIdx.x;
  if (i >= (E_DP << 4)) return;
  int e = i >> 4, c = i & 15;
  int s = src[e];
  int d = dst[e] - N_PROT;
  atomicAdd(&aggD[d * 16 + c], xp2[s * 16 + c]);
}

__global__ void k_divcnt(float* __restrict__ agg, const float* __restrict__ cnt,
                         int total, int shift) {
  int i = blockIdx.x * blockDim.x + threadIdx.x;
  if (i < total) agg[i] /= fmaxf(cnt[i >> shift], 1.0f);
}

// xd[:, :48] = emb / d_norm ; xd[:, 48:64] = xpd
__global__ void k_build_xd(const float* __restrict__ emb, const float* __restrict__ dnorm,
                           const float* __restrict__ xpd, float* __restrict__ xd) {
  int i = blockIdx.x * blockDim.x + threadIdx.x;
  if (i >= N_DRUG * RGCN_IN) return;
  int r = i >> 6, c = i & 63;
  xd[i] = (c < N_EMBED) ? emb[r * N_EMBED + c] / dnorm[r]
                        : xpd[r * PD_DIM + (c - N_EMBED)];
}

// Wt[t] = sum_b att[t,b] * basis[b]   (per = in*out elements per type)
__global__ void k_wt(const float* __restrict__ att, const float* __restrict__ basis,
                     float* __restrict__ Wt, int per) {
  int t = blockIdx.x;
  int idx = blockIdx.y * blockDim.x + threadIdx.x;
  if (idx >= per) return;
  float s = 0.f;
#pragma unroll 8
  for (int b = 0; b < N_BASE; ++b)
    s += att[t * N_BASE + b] * basis[b * per + idx];
  Wt[t * per + idx] = s;
}

// ---------------------------------------------------------------------------
// Big-workspace RGCN edge pass: acc[dst*C+c] += Y[etype][src*C+c]
// Y was produced by the batched WMMA GEMM; this is a pure L2 gather + atomic.
// thread = edge*channel (wave shares one edge -> coalesced Y row read).
// ---------------------------------------------------------------------------
__global__ void rgcn_gather(const float* __restrict__ Y, const int* __restrict__ src,
                            const int* __restrict__ dst, const int* __restrict__ et,
                            float* __restrict__ acc, int shift) {
  long i = (long)blockIdx.x * blockDim.x + threadIdx.x;
  if (i >= ((long)E_DD << shift)) return;
  int e = (int)(i >> shift);
  int c = (int)(i & ((1 << shift) - 1));
  // prefetch the streamed edge arrays well ahead (global_prefetch_b8)
  __builtin_prefetch(src + e + 4096, 0, 0);
  __builtin_prefetch(dst + e + 4096, 0, 0);
  __builtin_prefetch(et  + e + 4096, 0, 0);
  int t = et[e];
  int s = src[e], d = dst[e];
  atomicAdd(&acc[((long)d << shift) + c],
            Y[(((long)t * N_DRUG + s) << shift) + c]);
}

// ---------------------------------------------------------------------------
// Small-workspace fallback edge passes (LDS-cached W_t + shfl broadcast)
// ---------------------------------------------------------------------------
__global__ void rgcn_edge1(const float* __restrict__ x, const float* __restrict__ Wt,
                           const int* __restrict__ src, const int* __restrict__ dst,
                           const int* __restrict__ ranges, float* __restrict__ acc) {
  __shared__ float w[RGCN_IN * H1];  // 8 KB
  int t = blockIdx.x;
  for (int i = threadIdx.x; i < RGCN_IN * H1; i += blockDim.x)
    w[i] = Wt[t * RGCN_IN * H1 + i];
  __syncthreads();
  int start = ranges[2 * t], end = ranges[2 * t + 1];
  int lane  = threadIdx.x & 31;
  int wid   = blockIdx.y * (blockDim.x >> 5) + (threadIdx.x >> 5);
  int nw    = gridDim.y * (blockDim.x >> 5);
  for (int e = start + wid; e < end; e += nw) {
    int s = src[e], d = dst[e];
    float a0 = x[s * RGCN_IN + lane];
    float a1 = x[s * RGCN_IN + 32 + lane];
    float r = 0.f;
#pragma unroll
    for (int k = 0; k < 32; ++k) r += __shfl(a0, k) * w[k * H1 + lane];
#pragma unroll
    for (int k = 0; k < 32; ++k) r += __shfl(a1, k) * w[(32 + k) * H1 + lane];
    atomicAdd(&acc[d * H1 + lane], r);
  }
}

__global__ void rgcn_edge2(const float* __restrict__ x, const float* __restrict__ Wt,
                           const int* __restrict__ src, const int* __restrict__ dst,
                           const int* __restrict__ ranges, float* __restrict__ acc) {
  __shared__ float w[H1 * H2];  // 2 KB
  int t = blockIdx.x;
  for (int i = threadIdx.x; i < H1 * H2; i += blockDim.x)
    w[i] = Wt[t * H1 * H2 + i];
  __syncthreads();
  int start = ranges[2 * t], end = ranges[2 * t + 1];
  int lane  = threadIdx.x & 31;
  int c16   = lane & 15;
  int wid   = blockIdx.y * (blockDim.x >> 5) + (threadIdx.x >> 5);
  int nw    = gridDim.y * (blockDim.x >> 5);
  for (int e = start + wid; e < end; e += nw) {
    int s = src[e], d = dst[e];
    float a0 = x[s * H1 + lane];
    float r = 0.f;
#pragma unroll
    for (int k = 0; k < 32; ++k) r += __shfl(a0, k) * w[k * H2 + c16];
    if (lane < 16) atomicAdd(&acc[d * H2 + lane], r);
  }
}

// out = acc/max(cnt,1) + rt  (optional relu)
__global__ void k_combine(const float* __restrict__ acc, const float* __restrict__ rt,
                          const float* __restrict__ cnt, float* __restrict__ out,
                          int total, int shift, int dorelu) {
  int i = blockIdx.x * blockDim.x + threadIdx.x;
  if (i >= total) return;
  float v = acc[i] / fmaxf(cnt[i >> shift], 1.0f) + rt[i];
  out[i] = dorelu ? fmaxf(v, 0.f) : v;
}

// ---------------------------------------------------------------------------
extern "C" void kernel_launch(void* const* d_in, const int* in_sizes, int n_in,
                              void* d_out, int out_size, void* d_ws, size_t ws_size,
                              hipStream_t stream) {
  (void)in_sizes; (void)n_in; (void)out_size;

  const float* x_drug  = (const float*)d_in[0];
  const float* d_norm  = (const float*)d_in[1];
  const float* x_prot  = (const float*)d_in[2];
  const int*   pp      = (const int*)d_in[3];
  const int*   dp      = (const int*)d_in[4];
  const int*   dd      = (const int*)d_in[6];
  const int*   dd_et   = (const int*)d_in[7];
  const int*   dd_rng  = (const int*)d_in[8];
  const float* embed   = (const float*)d_in[9];
  const float* W1      = (const float*)d_in[10];
  const float* b1      = (const float*)d_in[11];
  const float* W2      = (const float*)d_in[12];
  const float* b2      = (const float*)d_in[13];
  const float* hg      = (const float*)d_in[14];
  const float* att1    = (const float*)d_in[15];
  const float* basis1  = (const float*)d_in[16];
  const float* root1   = (const float*)d_in[17];
  const float* att2    = (const float*)d_in[18];
  const float* basis2  = (const float*)d_in[19];
  const float* root2   = (const float*)d_in[20];
  float* out = (float*)d_out;

  const int *pp_src = pp, *pp_dst = pp + E_PP;
  const int *dp_src = dp, *dp_dst = dp + E_DP;
  const int *dd_src = dd, *dd_dst = dd + E_DD;

  // -------- workspace carve --------
  char* ws = (char*)d_ws;
  size_t off = 0;
  auto carve = [&](size_t nfloats) -> float* {
    float* p = (float*)(ws + off);
    off += (nfloats * sizeof(float) + 255) & ~(size_t)255;
    return p;
  };
  float* deg    = carve(N_PROT);
  float* dis    = carve(N_PROT);
  float* Ht1    = carve((size_t)N_PROT * H1);   // x_prot @ W1
  float* agg1   = carve((size_t)N_PROT * H1);   // -> xp1 (relu in place)
  float* Ht2    = carve((size_t)N_PROT * H2);   // xp1 @ W2
  float* agg2   = carve((size_t)N_PROT * H2);   // -> xp2
  float* aggD   = carve((size_t)N_DRUG * H2);
  float* cntD   = carve(N_DRUG);
  float* cntdd  = carve(N_DRUG);
  float* acc1   = carve((size_t)N_DRUG * H1);
  float* acc2   = carve((size_t)N_DRUG * H2);
  float* xpd    = carve((size_t)N_DRUG * PD_DIM);
  float* embout = carve((size_t)N_DRUG * N_EMBED);
  float* xd     = carve((size_t)N_DRUG * RGCN_IN);
  float* Wt1    = carve((size_t)N_ET * RGCN_IN * H1);
  float* Wt2    = carve((size_t)N_ET * H1 * H2);
  float* rt1    = carve((size_t)N_DRUG * H1);
  float* xd1    = carve((size_t)N_DRUG * H1);
  float* rt2    = carve((size_t)N_DRUG * H2);
  // Large per-type transformed features (L2-resident on MI455X: 98 MB < 192 MB)
  float* Y1     = carve((size_t)N_ET * N_DRUG * H1);   // 65.5 MB
  float* Y2     = carve((size_t)N_ET * N_DRUG * H2);   // 32.8 MB
  const bool bigpath = (off <= ws_size);   // deterministic path selection

  // -------- zero accumulators (graph-capturable async memsets) --------
  hipMemsetAsync(aggD,  0, (size_t)N_DRUG * H2 * sizeof(float), stream);
  hipMemsetAsync(cntD,  0, (size_t)N_DRUG * sizeof(float), stream);
  hipMemsetAsync(cntdd, 0, (size_t)N_DRUG * sizeof(float), stream);
  hipMemsetAsync(acc1,  0, (size_t)N_DRUG * H1 * sizeof(float), stream);
  hipMemsetAsync(acc2,  0, (size_t)N_DRUG * H2 * sizeof(float), stream);

  const int B = 256;
  auto nb = [](long n, int b) { return (unsigned)((n + b - 1) / b); };

  // -------- degrees for PP graph --------
  k_fill<<<nb(N_PROT, B), B, 0, stream>>>(deg, 1.0f, N_PROT);
  k_count<<<nb(E_PP, B), B, 0, stream>>>(pp_dst, deg, E_PP, 0);
  k_rsqrt<<<nb(N_PROT, B), B, 0, stream>>>(deg, dis, N_PROT);

  // -------- GCN layer 1: relu(gcn(x_prot, W1, b1)) --------
  gemm_wmma<<<dim3(N_PROT / 16, H1 / 16, 1), 32, 0, stream>>>(x_prot, W1, Ht1,
                                                              IN_PROT, H1, 0, 0);
  k_gcn_init<<<nb((long)N_PROT * H1, B), B, 0, stream>>>(Ht1, dis, b1, agg1, N_PROT * H1, 5);
  k_gcn_scatter<<<nb((long)E_PP << 5, B), B, 0, stream>>>(Ht1, dis, pp_src, pp_dst, agg1, E_PP, 5);
  k_relu<<<nb((long)N_PROT * H1, B), B, 0, stream>>>(agg1, N_PROT * H1);

  // -------- GCN layer 2 --------
  gemm_wmma<<<dim3(N_PROT / 16, H2 / 16, 1), 32, 0, stream>>>(agg1, W2, Ht2, H1, H2, 0, 0);
  k_gcn_init<<<nb((long)N_PROT * H2, B), B, 0, stream>>>(Ht2, dis, b2, agg2, N_PROT * H2, 4);
  k_gcn_scatter<<<nb((long)E_PP << 4, B), B, 0, stream>>>(Ht2, dis, pp_src, pp_dst, agg2, E_PP, 4);

  // -------- hierarchy conv: mean(prot->drug) @ hg_weight --------
  k_count<<<nb(E_DP, B), B, 0, stream>>>(dp_dst, cntD, E_DP, N_PROT);
  k_dp_scatter<<<nb((long)E_DP << 4, B), B, 0, stream>>>(agg2, dp_src, dp_dst, aggD);
  k_divcnt<<<nb((long)N_DRUG * H2, B), B, 0, stream>>>(aggD, cntD, N_DRUG * H2, 4);
  gemm_wmma<<<dim3(N_DRUG / 16, PD_DIM / 16, 1), 32, 0, stream>>>(aggD, hg, xpd,
                                                                  H2, PD_DIM, 0, 0);

  // -------- drug embedding + concat --------
  gemm_wmma<<<dim3(N_DRUG / 16, N_EMBED / 16, 1), 32, 0, stream>>>(x_drug, embed, embout,
                                                                   IN_DRUG, N_EMBED, 0, 0);
  k_build_xd<<<nb((long)N_DRUG * RGCN_IN, B), B, 0, stream>>>(embout, d_norm, xpd, xd);

  // -------- shared edge counts for dd graph --------
  k_count<<<nb(E_DD, B), B, 0, stream>>>(dd_dst, cntdd, E_DD, 0);

  // -------- RGCN layer 1 --------
  k_wt<<<dim3(N_ET, (RGCN_IN * H1) / B), B, 0, stream>>>(att1, basis1, Wt1, RGCN_IN * H1);
  if (bigpath) {
    // batched WMMA: Y1[t] = xd @ Wt1[t]  for all 128 types, then L2 gather
    gemm_wmma<<<dim3(N_DRUG / 16, H1 / 16, N_ET), 32, 0, stream>>>(
        xd, Wt1, Y1, RGCN_IN, H1, (long)RGCN_IN * H1, (long)N_DRUG * H1);
    rgcn_gather<<<nb((long)E_DD << 5, B), B, 0, stream>>>(Y1, dd_src, dd_dst, dd_et, acc1, 5);
  } else {
    rgcn_edge1<<<dim3(N_ET, 32), B, 0, stream>>>(xd, Wt1, dd_src, dd_dst, dd_rng, acc1);
  }
  gemm_wmma<<<dim3(N_DRUG / 16, H1 / 16, 1), 32, 0, stream>>>(xd, root1, rt1,
                                                              RGCN_IN, H1, 0, 0);
  k_combine<<<nb((long)N_DRUG * H1, B), B, 0, stream>>>(acc1, rt1, cntdd, xd1,
                                                        N_DRUG * H1, 5, 1);

  // -------- RGCN layer 2 --------
  k_wt<<<dim3(N_ET, (H1 * H2) / B), B, 0, stream>>>(att2, basis2, Wt2, H1 * H2);
  if (bigpath) {
    gemm_wmma<<<dim3(N_DRUG / 16, H2 / 16, N_ET), 32, 0, stream>>>(
        xd1, Wt2, Y2, H1, H2, (long)H1 * H2, (long)N_DRUG * H2);
    rgcn_gather<<<nb((long)E_DD << 4, B), B, 0, stream>>>(Y2, dd_src, dd_dst, dd_et, acc2, 4);
  } else {
    rgcn_edge2<<<dim3(N_ET, 32), B, 0, stream>>>(xd1, Wt2, dd_src, dd_dst, dd_rng, acc2);
  }
  gemm_wmma<<<dim3(N_DRUG / 16, H2 / 16, 1), 32, 0, stream>>>(xd1, root2, rt2,
                                                              H1, H2, 0, 0);
  k_combine<<<nb((long)N_DRUG * H2, B), B, 0, stream>>>(acc2, rt2, cntdd, out,
                                                        N_DRUG * H2, 4, 0);
}